// TripletLoss_78185584657057
// MI455X (gfx1250) — compile-verified
//
#include <hip/hip_runtime.h>
#include <hip/hip_bf16.h>
#include <math.h>

typedef __attribute__((ext_vector_type(2))) float v2f;
typedef __attribute__((ext_vector_type(4))) float v4f;
typedef __attribute__((ext_vector_type(8))) float v8f;

#define BATCH 4096
#define DIM   512
#define MARGIN 0.3f
#define EPSV   1e-12f
#define INF_BITS 0x7F800000u

// K-major LDS layout with xor-16 swizzle: conflict-free WMMA fragment reads.
#define LDSA_IDX(k, m) (((k) * 32) + ((m) ^ ((((k) >> 1) & 1) << 4)))

// ---------------------------------------------------------------------------
// Kernel 1: row squared norms (one wave per row) + init mining arrays + accs.
// ---------------------------------------------------------------------------
__global__ void __launch_bounds__(256)
norms_kernel(const float* __restrict__ emb, float* __restrict__ sq,
             unsigned* __restrict__ hp2g, unsigned* __restrict__ hn2g,
             float* __restrict__ accbuf) {
    const int gtid = (int)(blockIdx.x * blockDim.x + threadIdx.x);
    if (gtid < BATCH) { hp2g[gtid] = 0u; hn2g[gtid] = INF_BITS; }
    if (gtid < 2) accbuf[gtid] = 0.0f;

    const int wave = gtid >> 5;
    const int lane = gtid & 31;
    if (wave >= BATCH) return;
    const float* row = emb + (size_t)wave * DIM;
    float s = 0.0f;
#pragma unroll
    for (int j = 0; j < DIM / 32; ++j) {
        float v = row[lane + 32 * j];
        s = fmaf(v, v, s);
    }
#pragma unroll
    for (int off = 16; off > 0; off >>= 1) s += __shfl_xor(s, off, 32);
    if (lane == 0) sq[wave] = s;
}

// ---------------------------------------------------------------------------
// Kernel 2: fused Gram-GEMM (fp32 WMMA, 2x2 register blocking) + d^2 mining.
// Grid: (BATCH/32 row strips) x (2 column halves). 8 waves per workgroup.
// ---------------------------------------------------------------------------
__global__ void __launch_bounds__(256)
triplet_kernel(const float* __restrict__ emb, const int* __restrict__ labels,
               const float* __restrict__ sq,
               unsigned* __restrict__ hp2g, unsigned* __restrict__ hn2g) {
    __shared__ float ldsA[32 * DIM];   // 64 KB, K-major + swizzle

    const int row0 = blockIdx.x * 32;
    const int tid  = threadIdx.x;
    const int w    = tid >> 5;
    const int lane = tid & 31;
    const int half = lane >> 4;    // K sub-block / row half selector
    const int n    = lane & 15;    // tile column (and A-fragment row)

    // ---- Stage 32x512 A strip into LDS, transposed to K-major + swizzle ----
    {
        const v4f* src = (const v4f*)(emb + (size_t)row0 * DIM);
#pragma unroll
        for (int it = 0; it < 16; ++it) {
            const int i4 = tid + 256 * it;       // 4096 float4 total
            const int m  = i4 >> 7;              // row within strip (0..31)
            const int kb = (i4 & 127) << 2;      // k base (0..508)
            v4f vv = src[i4];
#pragma unroll
            for (int j = 0; j < 4; ++j)
                ldsA[LDSA_IDX(kb + j, m)] = vv[j];
        }
    }
    __syncthreads();

    // Per-lane row metadata: rows are row0 + mt*16 + v + 8*half.
    float sqr[2][8]; int lr[2][8];
#pragma unroll
    for (int mt = 0; mt < 2; ++mt)
#pragma unroll
        for (int v = 0; v < 8; ++v) {
            const int r = row0 + mt * 16 + v + 8 * half;
            sqr[mt][v] = sq[r];
            lr[mt][v]  = labels[r];
        }

    // Mining state on d^2 (all candidates >= EPSV > 0): identities 0 / +inf.
    float hp2[2][8], hn2[2][8];
#pragma unroll
    for (int mt = 0; mt < 2; ++mt)
#pragma unroll
        for (int v = 0; v < 8; ++v) { hp2[mt][v] = 0.0f; hn2[mt][v] = __builtin_inff(); }

    // Loop-invariant A fragment dword offsets (swizzle collapses to `half`).
    const int a0dw = 2 * half * 32 + (n + 16 * (0 ^ half));
    const int a1dw = 2 * half * 32 + (n + 16 * (1 ^ half));
    const float* __restrict__ pA0 = ldsA + a0dw;
    const float* __restrict__ pA1 = ldsA + a1dw;

    const int ctBase = blockIdx.y * 128;   // this half's first column tile

    for (int t = 0; t < 8; ++t) {
        const int ct0 = ctBase + 2 * (w + 8 * t);
        const int c0  = ct0 * 16;

        const float* bp0 = emb + (size_t)(c0 + n) * DIM + 2 * half;
        const float* bp1 = bp0 + (size_t)16 * DIM;

        v8f acc[2][2] = {};
#pragma unroll 4
        for (int k0 = 0; k0 < DIM; k0 += 4) {
            v2f a0; a0.x = pA0[k0 * 32]; a0.y = pA0[k0 * 32 + 32];  // ds_load_2addr_b32
            v2f a1; a1.x = pA1[k0 * 32]; a1.y = pA1[k0 * 32 + 32];
            v2f b0 = *(const v2f*)(bp0 + k0);                        // global_load_b64
            v2f b1 = *(const v2f*)(bp1 + k0);
            acc[0][0] = __builtin_amdgcn_wmma_f32_16x16x4_f32(false, a0, false, b0, (short)0, acc[0][0], false, false);
            acc[0][1] = __builtin_amdgcn_wmma_f32_16x16x4_f32(false, a0, false, b1, (short)0, acc[0][1], false, false);
            acc[1][0] = __builtin_amdgcn_wmma_f32_16x16x4_f32(false, a1, false, b0, (short)0, acc[1][0], false, false);
            acc[1][1] = __builtin_amdgcn_wmma_f32_16x16x4_f32(false, a1, false, b1, (short)0, acc[1][1], false, false);
        }

        // Branchless epilogue: fold both tiles into d^2 mining via selects.
#pragma unroll
        for (int nt = 0; nt < 2; ++nt) {
            const int   c   = (ct0 + nt) * 16 + n;
            const float sqc = sq[c];
            const int   lc  = labels[c];
#pragma unroll
            for (int mt = 0; mt < 2; ++mt)
#pragma unroll
                for (int v = 0; v < 8; ++v) {
                    const int r = row0 + mt * 16 + v + 8 * half;
                    const float d2 = fmaxf(sqr[mt][v] + sqc - 2.0f * acc[mt][nt][v], EPSV);
                    const bool same = (lc == lr[mt][v]);
                    const float pcand = (same && (r != c)) ? d2 : 0.0f;           // max identity
                    const float ncand = same ? __builtin_inff() : d2;             // min identity
                    hp2[mt][v] = fmaxf(hp2[mt][v], pcand);
                    hn2[mt][v] = fminf(hn2[mt][v], ncand);
                }
        }
    }

    // Reduce over the 16 columns held across each half-wave's lanes.
#pragma unroll
    for (int mt = 0; mt < 2; ++mt)
#pragma unroll
        for (int v = 0; v < 8; ++v) {
#pragma unroll
            for (int off = 8; off > 0; off >>= 1) {
                hp2[mt][v] = fmaxf(hp2[mt][v], __shfl_xor(hp2[mt][v], off, 16));
                hn2[mt][v] = fminf(hn2[mt][v], __shfl_xor(hn2[mt][v], off, 16));
            }
        }

    // Cross-workgroup combine: u32 atomics on positive-float bit patterns.
    if (n == 0) {
#pragma unroll
        for (int mt = 0; mt < 2; ++mt)
#pragma unroll
            for (int v = 0; v < 8; ++v) {
                const int r = row0 + mt * 16 + v + 8 * half;
                atomicMax(&hp2g[r], __float_as_uint(hp2[mt][v]));
                atomicMin(&hn2g[r], __float_as_uint(hn2[mt][v]));
            }
    }
}

// ---------------------------------------------------------------------------
// Kernel 3: per-row finalize (two sqrts per row) + global sum/count.
// ---------------------------------------------------------------------------
__global__ void __launch_bounds__(256)
reduce_kernel(const unsigned* __restrict__ hp2g, const unsigned* __restrict__ hn2g,
              float* __restrict__ accbuf) {
    __shared__ float sp[8], sc[8];
    const int r    = (int)(blockIdx.x * blockDim.x + threadIdx.x);
    const int w    = (int)(threadIdx.x >> 5);
    const int lane = (int)(threadIdx.x & 31);

    const float hp2 = __uint_as_float(hp2g[r]);
    const float hn2 = __uint_as_float(hn2g[r]);
    const bool valid = (hp2 > 0.0f) && (hn2 < __builtin_inff());
    float pr = 0.0f, cv = 0.0f;
    if (valid) {
        pr = fmaxf(__builtin_sqrtf(hp2) - __builtin_sqrtf(hn2) + MARGIN, 0.0f);
        cv = 1.0f;
    }
#pragma unroll
    for (int off = 16; off > 0; off >>= 1) {
        pr += __shfl_xor(pr, off, 32);
        cv += __shfl_xor(cv, off, 32);
    }
    if (lane == 0) { sp[w] = pr; sc[w] = cv; }
    __syncthreads();
    if (threadIdx.x == 0) {
        float S = 0.0f, C = 0.0f;
#pragma unroll
        for (int ww = 0; ww < 8; ++ww) { S += sp[ww]; C += sc[ww]; }
        atomicAdd(&accbuf[0], S);
        atomicAdd(&accbuf[1], C);
    }
}

// ---------------------------------------------------------------------------
// Kernel 4: scalar finalize.
// ---------------------------------------------------------------------------
__global__ void finalize_kernel(const float* __restrict__ accbuf,
                                float* __restrict__ out) {
    const float s = accbuf[0];
    const float c = accbuf[1];
    out[0] = (c > 0.0f) ? (s / fmaxf(c, 1.0f)) : 0.0f;
}

extern "C" void kernel_launch(void* const* d_in, const int* in_sizes, int n_in,
                              void* d_out, int out_size, void* d_ws, size_t ws_size,
                              hipStream_t stream) {
    const float* emb    = (const float*)d_in[0];   // [4096,512] fp32
    const int*   labels = (const int*)d_in[1];     // [4096] int32
    float*       out    = (float*)d_out;           // scalar fp32

    float*    sq     = (float*)d_ws;               // 4096 f32
    unsigned* hp2g   = (unsigned*)(sq + BATCH);    // 4096 u32 (f32 bits)
    unsigned* hn2g   = hp2g + BATCH;               // 4096 u32 (f32 bits)
    float*    accbuf = (float*)(hn2g + BATCH);     // {sum, count}

    norms_kernel<<<BATCH / 8, 256, 0, stream>>>(emb, sq, hp2g, hn2g, accbuf);

    dim3 grid(BATCH / 32, 2);
    triplet_kernel<<<grid, 256, 0, stream>>>(emb, labels, sq, hp2g, hn2g);

    reduce_kernel<<<BATCH / 256, 256, 0, stream>>>(hp2g, hn2g, accbuf);
    finalize_kernel<<<1, 1, 0, stream>>>(accbuf, out);
}